// NCellNSkipOld_11982958756685
// MI455X (gfx1250) — compile-verified
//
#include <hip/hip_runtime.h>
#include <hip/hip_bf16.h>

typedef __attribute__((ext_vector_type(2))) float v2f;
typedef __attribute__((ext_vector_type(4))) float v4f;
typedef __attribute__((ext_vector_type(8))) float v8f;

#define CCH   32
#define HWPIX 4096            // 64*64
#define NBATCH 64
#define NPIX  (NBATCH * HWPIX) // 262144
#define MROWS 128
#define KDIM  64

// ---------------------------------------------------------------------------
// Kernel 1: collapse the DARTS cell into Weff[128][64] = [A_n | B_n] per node,
// written to d_ws in WMMA A-fragment layout: ws[(kidx*8 + mtile)*64 + lane*2 + j]
// where for global (row,k): mtile=row>>4, kidx=k>>2, lane=(row&15)+16*((k>>1)&1), j=k&1.
// ---------------------------------------------------------------------------
__global__ __launch_bounds__(1024) void build_weff(
    const float* __restrict__ Wp0, const float* __restrict__ Wp1,
    const float* __restrict__ Wedges, const float* __restrict__ alphas,
    float* __restrict__ ws)
{
    __shared__ float sA[6][32][32];   // per-state matrix applied to x0
    __shared__ float sB[6][32][32];   // per-state matrix applied to x1
    __shared__ float wmix[32][32];    // current edge's softmax-mixed matrix

    const int tid = threadIdx.x;
    const int o = tid >> 5;
    const int c = tid & 31;

    sA[0][o][c] = Wp0[o * 32 + c];  sB[0][o][c] = 0.0f;
    sA[1][o][c] = 0.0f;             sB[1][o][c] = Wp1[o * 32 + c];
    __syncthreads();

    int offset = 0;
    for (int node = 0; node < 4; ++node) {
        const int nst = 2 + node;          // incoming edges / #states so far
        float accA = 0.0f, accB = 0.0f;
        for (int j = 0; j < nst; ++j) {
            const int e = offset + j;
            // softmax over 5 ops (redundant per thread, trivial cost)
            const float a0 = alphas[e * 5 + 0];
            const float a1 = alphas[e * 5 + 1];
            const float a2 = alphas[e * 5 + 2];
            const float a3 = alphas[e * 5 + 3];
            const float a4 = alphas[e * 5 + 4];
            float mx = fmaxf(fmaxf(fmaxf(a0, a1), fmaxf(a2, a3)), a4);
            const float e0 = __expf(a0 - mx), e1 = __expf(a1 - mx),
                        e2 = __expf(a2 - mx), e3 = __expf(a3 - mx),
                        e4 = __expf(a4 - mx);
            const float inv = 1.0f / (e0 + e1 + e2 + e3 + e4);
            const float* We = Wedges + (size_t)e * 5 * 1024 + (o * 32 + c);
            wmix[o][c] = (e0 * We[0] + e1 * We[1024] + e2 * We[2048] +
                          e3 * We[3072] + e4 * We[4096]) * inv;
            __syncthreads();
            // acc += wmix @ state_j   (element (o,c))
            float ra = 0.0f, rb = 0.0f;
            #pragma unroll
            for (int t = 0; t < 32; ++t) {
                const float wv = wmix[o][t];
                ra = fmaf(wv, sA[j][t][c], ra);
                rb = fmaf(wv, sB[j][t][c], rb);
            }
            accA += ra;  accB += rb;
            __syncthreads();               // before wmix is overwritten
        }
        sA[nst][o][c] = accA;  sB[nst][o][c] = accB;
        __syncthreads();
        offset += nst;
    }

    // Emit Weff (rows = node*32+o, cols 0..31 from A, 32..63 from B) as fragments.
    for (int idx = tid; idx < MROWS * KDIM; idx += 1024) {
        const int row  = idx >> 6;
        const int k    = idx & 63;
        const int node = row >> 5;
        const int oo   = row & 31;
        const float v  = (k < 32) ? sA[2 + node][oo][k] : sB[2 + node][oo][k - 32];
        const int mtile = row >> 4;
        const int kidx  = k >> 2;
        const int half  = (k >> 1) & 1;
        const int jj    = k & 1;
        const int lane  = (row & 15) + 16 * half;
        ws[(kidx * 8 + mtile) * 64 + lane * 2 + jj] = v;
    }
}

// ---------------------------------------------------------------------------
// Kernel 2: Out(128 x N) = Weff(128x64) @ [X0;X1](64 x N), N = 262144 pixels.
// 8 waves/block; each wave: one 16-pixel tile, full M=128 via 8 f32 WMMA accs.
// ---------------------------------------------------------------------------
__global__ __launch_bounds__(256) void darts_gemm(
    const float* __restrict__ x0, const float* __restrict__ x1,
    const float* __restrict__ wfrag, float* __restrict__ out)
{
    __shared__ float lA[MROWS * KDIM];          // 32 KB fragment table

    const int tid = threadIdx.x;
    {   // cooperative b128 load of the fragment table
        const v4f* src = (const v4f*)wfrag;
        v4f* dst = (v4f*)lA;
        #pragma unroll
        for (int i = 0; i < 8; ++i) dst[tid + 256 * i] = src[tid + 256 * i];
    }
    __syncthreads();

    const int wv   = tid >> 5;
    const int lane = tid & 31;
    const int half = lane >> 4;                 // 0: lanes 0-15, 1: lanes 16-31
    const int n    = lane & 15;                 // column within 16-pixel tile

    const int tile   = blockIdx.x * 8 + wv;     // 0 .. 16383
    const int pixel0 = tile * 16;               // HW=4096 is 16-aligned: no straddle
    const int b      = pixel0 >> 12;
    const int pin    = pixel0 & 4095;

    const float* p0 = x0 + (size_t)b * CCH * HWPIX + pin + n;
    const float* p1 = x1 + (size_t)b * CCH * HWPIX + pin + n;

    v8f acc[8];
    #pragma unroll
    for (int m = 0; m < 8; ++m) { v8f z = {}; acc[m] = z; }

    #pragma unroll
    for (int kidx = 0; kidx < 16; ++kidx) {
        const int k0 = kidx * 4;
        const int kk = k0 + 2 * half;           // B rows kk, kk+1 for this lane
        v2f bfrag;
        if (k0 < 32) {                          // wave-uniform branch: EXEC stays full
            bfrag.x = p0[(size_t)kk * HWPIX];
            bfrag.y = p0[(size_t)(kk + 1) * HWPIX];
        } else {
            bfrag.x = p1[(size_t)(kk - 32) * HWPIX];
            bfrag.y = p1[(size_t)(kk - 31) * HWPIX];
        }
        const float* abase = &lA[(kidx * 8) * 64 + lane * 2];
        #pragma unroll
        for (int m = 0; m < 8; ++m) {
            v2f afrag = *(const v2f*)(abase + m * 64);   // ds_load_b64, conflict-free
            acc[m] = __builtin_amdgcn_wmma_f32_16x16x4_f32(
                /*neg_a=*/false, afrag, /*neg_b=*/false, bfrag,
                /*c_mod=*/(short)0, acc[m], /*reuse_a=*/false, /*reuse_b=*/false);
        }
    }

    // C/D layout: VGPR r -> rows M=r (lanes 0-15) and M=r+8 (lanes 16-31)
    float* ob = out + (size_t)b * MROWS * HWPIX + pin + n;
    #pragma unroll
    for (int m = 0; m < 8; ++m) {
        #pragma unroll
        for (int r = 0; r < 8; ++r) {
            ob[(size_t)(m * 16 + half * 8 + r) * HWPIX] = acc[m][r];
        }
    }
}

extern "C" void kernel_launch(void* const* d_in, const int* in_sizes, int n_in,
                              void* d_out, int out_size, void* d_ws, size_t ws_size,
                              hipStream_t stream) {
    const float* input0 = (const float*)d_in[0];
    const float* input1 = (const float*)d_in[1];
    const float* Wp0    = (const float*)d_in[2];
    const float* Wp1    = (const float*)d_in[3];
    const float* Wedges = (const float*)d_in[4];
    const float* alphas = (const float*)d_in[5];
    float* ws  = (float*)d_ws;
    float* out = (float*)d_out;

    build_weff<<<1, 1024, 0, stream>>>(Wp0, Wp1, Wedges, alphas, ws);
    darts_gemm<<<NPIX / (16 * 8), 256, 0, stream>>>(input0, input1, ws, out);
}